// GBottleneck_85160611545813
// MI455X (gfx1250) — compile-verified
//
#include <hip/hip_runtime.h>

// GCN bottleneck pipeline for MI455X (gfx1250, wave32).
// Dense GEMMs use V_WMMA_F32_16X16X4_F32 (fp32-exact; GEMM compute is far
// below the SpMM scatter-atomic cost, so higher-rate low-precision WMMA buys
// nothing). All N x H intermediates (76.8 MB) are L2-resident (192 MB L2).

typedef __attribute__((ext_vector_type(2))) float v2f;
typedef __attribute__((ext_vector_type(8))) float v8f;

constexpr int HD  = 192;      // hidden width
constexpr int NN  = 100000;   // nodes
constexpr int EE  = 1600000;  // edges

// ---------------------------------------------------------------------------
// Dual GEMM:  SUP = A @ W
//             C2  = A @ LW + bias                (RES == nullptr)
//             C2  = 0.5 * (RES + A @ LW + bias)  (RES != nullptr)
// A: [M x K] row-major, W/LW: [K x HD] row-major.
// Block = 256 threads = 8 waves; tile = 32 rows x 192 cols.
// Wave: rowHalf (0..1) x colGrp (0..3) -> 16 rows x 48 cols (3 WMMA tiles).
// WMMA f32 16x16x4 operand layout (ISA 7.12.2):
//   A: lane m%16 holds row m; VGPR0/1 = K0/K1 (lanes 0-15), K2/K3 (16-31)
//   B: lane n%16 holds col n; VGPR0/1 = K0/K1 (lanes 0-15), K2/K3 (16-31)
//   D: VGPR j -> row j (lanes 0-15) / row 8+j (lanes 16-31); lane%16 -> col
// ---------------------------------------------------------------------------
__global__ __launch_bounds__(256)
void dual_gemm_wmma(const float* __restrict__ A,
                    const float* __restrict__ W,
                    const float* __restrict__ LW,
                    const float* __restrict__ bias,
                    const float* RES,       // may alias C2 element-wise (safe)
                    float* __restrict__ SUP,
                    float* C2,
                    int K)
{
    const int lane    = threadIdx.x & 31;
    const int wid     = threadIdx.x >> 5;
    const int rowHalf = wid >> 2;               // 0..1
    const int colGrp  = wid & 3;                // 0..3
    const int mBase   = blockIdx.x * 32 + rowHalf * 16;
    const int nB      = colGrp * 48;
    const int l16     = lane & 15;
    const int hi      = lane >> 4;              // 0 or 1
    const int koff    = hi * 2;                 // K sub-offset for this half

    const float* Arow = A + (size_t)(mBase + l16) * K;

    const v8f z = {0.f, 0.f, 0.f, 0.f, 0.f, 0.f, 0.f, 0.f};
    v8f acc1[3] = {z, z, z};
    v8f acc2[3] = {z, z, z};

    for (int k0 = 0; k0 < K; k0 += 4) {
        // A fragment: single 8B load per lane (K even, so 8B aligned)
        const float2 av = *reinterpret_cast<const float2*>(Arow + k0 + koff);
        v2f af; af.x = av.x; af.y = av.y;

        const float* wrow = W  + (size_t)(k0 + koff) * HD + nB + l16;
        const float* lrow = LW + (size_t)(k0 + koff) * HD + nB + l16;
#pragma unroll
        for (int t = 0; t < 3; ++t) {
            v2f bw; bw.x = wrow[t * 16]; bw.y = wrow[t * 16 + HD];
            v2f bl; bl.x = lrow[t * 16]; bl.y = lrow[t * 16 + HD];
            acc1[t] = __builtin_amdgcn_wmma_f32_16x16x4_f32(
                false, af, false, bw, (short)0, acc1[t], false, false);
            acc2[t] = __builtin_amdgcn_wmma_f32_16x16x4_f32(
                false, af, false, bl, (short)0, acc2[t], false, false);
        }
    }

#pragma unroll
    for (int t = 0; t < 3; ++t) {
        const int n = nB + t * 16 + l16;
        const float bv = bias[n];
#pragma unroll
        for (int j = 0; j < 8; ++j) {
            const int row = mBase + hi * 8 + j;
            const size_t idx = (size_t)row * HD + n;
            SUP[idx] = acc1[t][j];
            float v = acc2[t][j] + bv;
            if (RES) v = 0.5f * (RES[idx] + v);   // residual fold; spmm adds 0.5*agg
            C2[idx] = v;
        }
    }
}

// ---------------------------------------------------------------------------
// SpMM scatter: OUT[dst[e]] += scale * ew[e] * SUP[src[e]]
// One thread per (edge, float4 chunk): 16B gather + 4 fp32 atomics (L2).
// ---------------------------------------------------------------------------
__global__ __launch_bounds__(256)
void spmm_scatter(const float* __restrict__ SUP, const int* __restrict__ srcI,
                  const int* __restrict__ dstI, const float* __restrict__ ew,
                  float* OUT, float scale, int E)
{
    constexpr int CH = HD / 4;  // 48 float4 chunks per row
    const long long tid = (long long)blockIdx.x * 256 + threadIdx.x;
    if (tid >= (long long)E * CH) return;
    const int e = (int)(tid / CH);
    const int c = (int)(tid - (long long)e * CH);
    const float w = ew[e] * scale;
    const float4 v = *reinterpret_cast<const float4*>(
        SUP + (size_t)srcI[e] * HD + c * 4);
    float* o = OUT + (size_t)dstI[e] * HD + c * 4;
    atomicAdd(o + 0, v.x * w);
    atomicAdd(o + 1, v.y * w);
    atomicAdd(o + 2, v.z * w);
    atomicAdd(o + 3, v.w * w);
}

// ---------------------------------------------------------------------------
// conv2 stage 1 (OUT=3): SUP2 = h @ w2 ; OUT = h @ l2 + b2.
// Weights (192x3) staged in LDS; one thread per node.
// ---------------------------------------------------------------------------
__global__ __launch_bounds__(256)
void conv2_node(const float* __restrict__ Hm, const float* __restrict__ W2,
                const float* __restrict__ L2, const float* __restrict__ B2,
                float* __restrict__ SUP2, float* __restrict__ OUT)
{
    __shared__ float w2s[HD * 3];
    __shared__ float l2s[HD * 3];
    for (int i = threadIdx.x; i < HD * 3; i += 256) {
        w2s[i] = W2[i];
        l2s[i] = L2[i];
    }
    __syncthreads();

    const int n = blockIdx.x * 256 + threadIdx.x;
    if (n >= NN) return;
    const float* hp = Hm + (size_t)n * HD;
    float s0 = 0.f, s1 = 0.f, s2 = 0.f, p0 = 0.f, p1 = 0.f, p2 = 0.f;
    for (int k = 0; k < HD; ++k) {
        const float hv = hp[k];
        s0 += hv * w2s[k * 3 + 0]; s1 += hv * w2s[k * 3 + 1]; s2 += hv * w2s[k * 3 + 2];
        p0 += hv * l2s[k * 3 + 0]; p1 += hv * l2s[k * 3 + 1]; p2 += hv * l2s[k * 3 + 2];
    }
    SUP2[(size_t)n * 3 + 0] = s0;
    SUP2[(size_t)n * 3 + 1] = s1;
    SUP2[(size_t)n * 3 + 2] = s2;
    OUT[(size_t)n * 3 + 0] = p0 + B2[0];
    OUT[(size_t)n * 3 + 1] = p1 + B2[1];
    OUT[(size_t)n * 3 + 2] = p2 + B2[2];
}

// conv2 stage 2: per-edge scatter of the 3 output features.
__global__ __launch_bounds__(256)
void spmm3_scatter(const float* __restrict__ SUP2, const int* __restrict__ srcI,
                   const int* __restrict__ dstI, const float* __restrict__ ew,
                   float* OUT, int E)
{
    const int e = blockIdx.x * 256 + threadIdx.x;
    if (e >= E) return;
    const float w = ew[e];
    const int s = srcI[e], d = dstI[e];
    atomicAdd(&OUT[(size_t)d * 3 + 0], SUP2[(size_t)s * 3 + 0] * w);
    atomicAdd(&OUT[(size_t)d * 3 + 1], SUP2[(size_t)s * 3 + 1] * w);
    atomicAdd(&OUT[(size_t)d * 3 + 2], SUP2[(size_t)s * 3 + 2] * w);
}

// ---------------------------------------------------------------------------
extern "C" void kernel_launch(void* const* d_in, const int* in_sizes, int n_in,
                              void* d_out, int out_size, void* d_ws, size_t ws_size,
                              hipStream_t stream)
{
    (void)in_sizes; (void)n_in; (void)out_size; (void)ws_size;
    constexpr int N = NN, E = EE, IND = 960, NBK = 6;

    const float* x   = (const float*)d_in[0];
    const int*   src = (const int*)  d_in[1];
    const int*   dst = (const int*)  d_in[2];
    const float* ew  = (const float*)d_in[3];
    const float* w1  = (const float*)d_in[4];
    const float* l1  = (const float*)d_in[5];
    const float* b1  = (const float*)d_in[6];
    const float* wb  = (const float*)d_in[7];
    const float* lb  = (const float*)d_in[8];
    const float* bb  = (const float*)d_in[9];
    const float* w2  = (const float*)d_in[10];
    const float* l2  = (const float*)d_in[11];
    const float* b2  = (const float*)d_in[12];

    float* xout = (float*)d_out;                 // [N,3]
    float* xcat = xout + (size_t)N * 3;          // [N,HD]

    // Workspace: 2 x N*HD floats (153.6 MB). x_cat region doubles as the
    // per-block 'a' scratch until the final d2d publish.
    float* hbuf = (float*)d_ws;                  // current hidden h
    float* sup  = hbuf + (size_t)N * HD;         // A @ W staging (also SUP2)
    float* abuf = xcat;                          // first-gcn output per block

    const dim3 blk(256);
    const int gemmBlocks = N / 32;                                  // 3125
    const long long spmmTotal = (long long)E * (HD / 4);
    const int spmmBlocks = (int)((spmmTotal + 255) / 256);          // 300000

    // conv1: sup = x@w1 ; h = x@l1 + b1 ; then h += agg(sup)
    dual_gemm_wmma<<<gemmBlocks, blk, 0, stream>>>(x, w1, l1, b1, nullptr,
                                                   sup, hbuf, IND);
    spmm_scatter<<<spmmBlocks, blk, 0, stream>>>(sup, src, dst, ew, hbuf, 1.0f, E);

    // 6 GResBlocks, residual scale folded: h = 0.5*h + 0.5*(loop+b) + 0.5*agg
    for (int i = 0; i < NBK; ++i) {
        const float* wA = wb + (size_t)(2 * i) * HD * HD;
        const float* lA = lb + (size_t)(2 * i) * HD * HD;
        const float* bA = bb + (size_t)(2 * i) * HD;
        dual_gemm_wmma<<<gemmBlocks, blk, 0, stream>>>(hbuf, wA, lA, bA, nullptr,
                                                       sup, abuf, HD);
        spmm_scatter<<<spmmBlocks, blk, 0, stream>>>(sup, src, dst, ew, abuf, 1.0f, E);

        const float* wB = wb + (size_t)(2 * i + 1) * HD * HD;
        const float* lB = lb + (size_t)(2 * i + 1) * HD * HD;
        const float* bB = bb + (size_t)(2 * i + 1) * HD;
        dual_gemm_wmma<<<gemmBlocks, blk, 0, stream>>>(abuf, wB, lB, bB, hbuf,
                                                       sup, hbuf, HD);
        spmm_scatter<<<spmmBlocks, blk, 0, stream>>>(sup, src, dst, ew, hbuf, 0.5f, E);
    }

    // Publish x_cat (capture-safe d2d copy).
    hipMemcpyAsync(xcat, hbuf, (size_t)N * HD * sizeof(float),
                   hipMemcpyDeviceToDevice, stream);

    // conv2: x_out = agg(h@w2) + h@l2 + b2
    conv2_node<<<(N + 255) / 256, blk, 0, stream>>>(hbuf, w2, l2, b2, sup, xout);
    spmm3_scatter<<<(E + 255) / 256, blk, 0, stream>>>(sup, src, dst, ew, xout, E);
}